// ASTAutoencoder_22565758173969
// MI455X (gfx1250) — compile-verified
//
#include <hip/hip_runtime.h>
#include <hip/hip_bf16.h>

// ---------------------------------------------------------------------------
// GCN autoencoder for MI455X (gfx1250, wave32).
//   Dense GEMMs   : V_WMMA_F32_16X16X4_F32 (fp32 in/out, 16x16 tile, K step 4)
//   Encoder aggr  : edge-driven gather + global_atomic_add_f32 (L2-resident)
//   Decoder aggr  : analytic +/-1,+/-2 stencil, degrees computed in closed form
// ---------------------------------------------------------------------------

typedef float v2f __attribute__((ext_vector_type(2)));
typedef float v8f __attribute__((ext_vector_type(8)));

#define NPER 100   // nodes per graph (reference constant)

// ---------------------------------------------------------------------------
// WMMA GEMM: Y[M x NOUT] = X[M x K] @ W[K x NOUT] (+bias) (+relu)
// One wave per 16-row tile, 4 waves per block. K, NOUT multiples of 4/16.
// ---------------------------------------------------------------------------
template<int K, int NOUT, bool BIAS, bool RELU>
__global__ __launch_bounds__(128)
void gemm_wmma(const float* __restrict__ X, const float* __restrict__ W,
               const float* __restrict__ bias, float* __restrict__ Y, int M) {
  const int wave = threadIdx.x >> 5;
  const int lane = threadIdx.x & 31;
  const long tile = (long)blockIdx.x * 4 + wave;
  const long row0 = tile * 16;
  if (row0 >= M) return;                 // wave-uniform: EXEC stays all-1s

  constexpr int NT = NOUT / 16;
  v8f acc[NT] = {};

  const int lrow  = lane & 15;           // row within 16 (A) / col within 16 (B)
  const int khalf = (lane >> 4) << 1;    // 0 for lanes 0-15, 2 for lanes 16-31
  const float* __restrict__ xrow = X + (row0 + lrow) * (long)K;

  for (int k = 0; k < K; k += 4) {
    v2f a;                               // A 16x4 fp32 fragment
    a.x = xrow[k + khalf];
    a.y = xrow[k + khalf + 1];
#pragma unroll
    for (int t = 0; t < NT; ++t) {
      const float* wp = W + (long)(k + khalf) * NOUT + t * 16 + lrow;
      v2f b;                             // B 4x16 fp32 fragment
      b.x = wp[0];
      b.y = wp[NOUT];
      acc[t] = __builtin_amdgcn_wmma_f32_16x16x4_f32(
          false, a, false, b, (short)0, acc[t], false, false);
    }
  }

  const int rbase = (lane >> 4) * 8;     // lanes 0-15: rows 0-7; 16-31: rows 8-15
#pragma unroll
  for (int t = 0; t < NT; ++t) {
    const int col = t * 16 + lrow;
    float bv = BIAS ? bias[col] : 0.0f;
#pragma unroll
    for (int r = 0; r < 8; ++r) {
      float v = acc[t][r] + bv;
      if (RELU) v = fmaxf(v, 0.0f);
      Y[(row0 + rbase + r) * (long)NOUT + col] = v;
    }
  }
}

// ---------------------------------------------------------------------------
// Elementwise / sparse helper kernels
// ---------------------------------------------------------------------------
__global__ void k_zero(float* __restrict__ p, long n) {
  long i = (long)blockIdx.x * blockDim.x + threadIdx.x;
  if (i < n) p[i] = 0.0f;
}

__global__ void k_count(const int* __restrict__ dst, float* __restrict__ deg, int E) {
  int e = blockIdx.x * blockDim.x + threadIdx.x;
  if (e < E) unsafeAtomicAdd(&deg[dst[e]], 1.0f);
}

__global__ void k_dinv(float* __restrict__ deg, int n) {     // in place deg -> rsqrt(deg+1)
  int i = blockIdx.x * blockDim.x + threadIdx.x;
  if (i < n) deg[i] = rsqrtf(deg[i] + 1.0f);                 // +1 = self loop
}

// agg[dst] += h[src] * dinv[src]*dinv[dst]; one lane handles 2 features of one edge
__global__ void k_scatter(const int* __restrict__ src, const int* __restrict__ dst,
                          const float* __restrict__ dinv, const float* __restrict__ h,
                          float* __restrict__ agg, int E) {
  long gid = (long)blockIdx.x * blockDim.x + threadIdx.x;
  int e = (int)(gid >> 5);
  if (e >= E) return;
  int f = (int)(gid & 31) * 2;
  int s = src[e], d = dst[e];
  float nrm = dinv[s] * dinv[d];
  const float2 hv = *(const float2*)(h + (long)s * 64 + f);
  unsafeAtomicAdd(&agg[(long)d * 64 + f],     hv.x * nrm);
  unsafeAtomicAdd(&agg[(long)d * 64 + f + 1], hv.y * nrm);
}

// out = (relu?)(agg + h*dinv^2 + bias)   (in place on agg)
template<bool RELU>
__global__ void k_combine(const float* __restrict__ h, const float* __restrict__ dinv,
                          const float* __restrict__ bias, float* __restrict__ agg, long n64) {
  long gid = (long)blockIdx.x * blockDim.x + threadIdx.x;
  if (gid >= n64) return;
  int i = (int)(gid >> 6), f = (int)(gid & 63);
  float di = dinv[i];
  float v = agg[gid] + h[gid] * di * di + bias[f];
  agg[gid] = RELU ? fmaxf(v, 0.0f) : v;
}

// mean pool: emb[g][f] = mean over NPER consecutive nodes
__global__ void k_pool(const float* __restrict__ h, float* __restrict__ emb, int ng) {
  int gid = blockIdx.x * blockDim.x + threadIdx.x;
  if (gid >= ng * 64) return;
  int g = gid >> 6, f = gid & 63;
  const float* p = h + (long)g * NPER * 64 + f;
  float s = 0.0f;
  for (int i = 0; i < NPER; ++i) s += p[i * 64];
  emb[gid] = s * (1.0f / NPER);
}

__global__ void k_broadcast(const float* __restrict__ init, float* __restrict__ z, long n64) {
  long gid = (long)blockIdx.x * blockDim.x + threadIdx.x;
  if (gid >= n64) return;
  long node = gid >> 6;
  z[gid] = init[(node / NPER) * 64 + (gid & 63)];
}

// decoder chain-edge degree: neighbors at local-2,-1,+1,+2 in [0,NPER) plus self loop
__device__ __forceinline__ float dec_dinv(int local) {
  int c = (local >= 2) + (local >= 1) + (local <= NPER - 2) + (local <= NPER - 3);
  return rsqrtf((float)(c + 1));
}

// decoder GCN aggregation as a stencil: no edge list, no atomics
template<bool RELU>
__global__ void k_dec_agg(const float* __restrict__ h, const float* __restrict__ bias,
                          float* __restrict__ out, long n64) {
  long gid = (long)blockIdx.x * blockDim.x + threadIdx.x;
  if (gid >= n64) return;
  int i = (int)(gid >> 6), f = (int)(gid & 63);
  int local = i % NPER;
  float dj = dec_dinv(local);
  float s = h[gid] * dj * dj;                                   // self loop
  if (local >= 1)        s += h[gid - 64]  * dec_dinv(local - 1) * dj;
  if (local >= 2)        s += h[gid - 128] * dec_dinv(local - 2) * dj;
  if (local <= NPER - 2) s += h[gid + 64]  * dec_dinv(local + 1) * dj;
  if (local <= NPER - 3) s += h[gid + 128] * dec_dinv(local + 2) * dj;
  s += bias[f];
  out[gid] = RELU ? fmaxf(s, 0.0f) : s;
}

// ---------------------------------------------------------------------------
// Orchestration
// ---------------------------------------------------------------------------
static inline long cdiv(long a, long b) { return (a + b - 1) / b; }

extern "C" void kernel_launch(void* const* d_in, const int* in_sizes, int n_in,
                              void* d_out, int out_size, void* d_ws, size_t ws_size,
                              hipStream_t stream) {
  const float* x      = (const float*)d_in[0];     // [N,128]
  const int*   ei     = (const int*)  d_in[1];     // [2,E]
  const float* enc_W1 = (const float*)d_in[3];
  const float* enc_b1 = (const float*)d_in[4];
  const float* enc_W2 = (const float*)d_in[5];
  const float* enc_b2 = (const float*)d_in[6];
  const float* enc_W3 = (const float*)d_in[7];
  const float* enc_b3 = (const float*)d_in[8];
  const float* dec_Wt = (const float*)d_in[9];
  const float* dec_bt = (const float*)d_in[10];
  const float* dec_W1 = (const float*)d_in[11];
  const float* dec_b1 = (const float*)d_in[12];
  const float* dec_W2 = (const float*)d_in[13];
  const float* dec_b2 = (const float*)d_in[14];
  const float* dec_W3 = (const float*)d_in[15];
  const float* dec_b3 = (const float*)d_in[16];
  const float* out_W  = (const float*)d_in[17];
  const float* out_b  = (const float*)d_in[18];

  const int N  = in_sizes[0] / 128;                // 200000
  const int E  = in_sizes[1] / 2;                  // 3200000
  const int NG = N / NPER;                         // 2000
  const long n64 = (long)N * 64;

  const int* src = ei;
  const int* dst = ei + E;

  // workspace layout (floats)
  float* bufA = (float*)d_ws;                      // N*64
  float* bufB = bufA + n64;                        // N*64
  float* dinv = bufB + n64;                        // N
  float* emb  = dinv + N;                          // NG*64
  float* initb = emb + (long)NG * 64;              // NG*64

  const int T = 256;
  dim3 bT(T);

  // ---- encoder degree -> dinv (dinv = rsqrt(deg+1)) ----
  k_zero<<<cdiv(N, T), bT, 0, stream>>>(dinv, N);
  k_count<<<cdiv(E, T), bT, 0, stream>>>(dst, dinv, E);
  k_dinv<<<cdiv(N, T), bT, 0, stream>>>(dinv, N);

  const long gemmN  = cdiv(N, 64);                 // 4 waves/block, 16 rows/wave
  const long gemmNG = cdiv(NG, 64);
  const long scatG  = cdiv((long)E * 32, T);
  const long elemG  = cdiv(n64, T);

  // ---- encoder layer 1: h = x@W1 ; agg ; relu ----
  gemm_wmma<128, 64, false, false><<<gemmN, 128, 0, stream>>>(x, enc_W1, nullptr, bufA, N);
  k_zero<<<elemG, bT, 0, stream>>>(bufB, n64);
  k_scatter<<<scatG, bT, 0, stream>>>(src, dst, dinv, bufA, bufB, E);
  k_combine<true><<<elemG, bT, 0, stream>>>(bufA, dinv, enc_b1, bufB, n64);

  // ---- encoder layer 2 ----
  gemm_wmma<64, 64, false, false><<<gemmN, 128, 0, stream>>>(bufB, enc_W2, nullptr, bufA, N);
  k_zero<<<elemG, bT, 0, stream>>>(bufB, n64);
  k_scatter<<<scatG, bT, 0, stream>>>(src, dst, dinv, bufA, bufB, E);
  k_combine<true><<<elemG, bT, 0, stream>>>(bufA, dinv, enc_b2, bufB, n64);

  // ---- encoder layer 3 (no relu) ----
  gemm_wmma<64, 64, false, false><<<gemmN, 128, 0, stream>>>(bufB, enc_W3, nullptr, bufA, N);
  k_zero<<<elemG, bT, 0, stream>>>(bufB, n64);
  k_scatter<<<scatG, bT, 0, stream>>>(src, dst, dinv, bufA, bufB, E);
  k_combine<false><<<elemG, bT, 0, stream>>>(bufA, dinv, enc_b3, bufB, n64);

  // ---- pool + decoder init + broadcast ----
  k_pool<<<cdiv((long)NG * 64, T), bT, 0, stream>>>(bufB, emb, NG);
  gemm_wmma<64, 64, true, false><<<gemmNG, 128, 0, stream>>>(emb, dec_Wt, dec_bt, initb, NG);
  k_broadcast<<<elemG, bT, 0, stream>>>(initb, bufA, n64);

  // ---- decoder GCN layers (stencil aggregation) ----
  gemm_wmma<64, 64, false, false><<<gemmN, 128, 0, stream>>>(bufA, dec_W1, nullptr, bufB, N);
  k_dec_agg<true><<<elemG, bT, 0, stream>>>(bufB, dec_b1, bufA, n64);

  gemm_wmma<64, 64, false, false><<<gemmN, 128, 0, stream>>>(bufA, dec_W2, nullptr, bufB, N);
  k_dec_agg<true><<<elemG, bT, 0, stream>>>(bufB, dec_b2, bufA, n64);

  gemm_wmma<64, 64, false, false><<<gemmN, 128, 0, stream>>>(bufA, dec_W3, nullptr, bufB, N);
  k_dec_agg<true><<<elemG, bT, 0, stream>>>(bufB, dec_b3, bufA, n64);

  // ---- output projection straight into d_out ----
  gemm_wmma<64, 128, true, false><<<gemmN, 128, 0, stream>>>(bufA, out_W, out_b,
                                                             (float*)d_out, N);
}